// CompRes_10849087389983
// MI455X (gfx1250) — compile-verified
//
#include <hip/hip_runtime.h>
#include <hip/hip_bf16.h>
#include <math.h>

typedef __attribute__((ext_vector_type(2))) float v2f;
typedef __attribute__((ext_vector_type(8))) float v8f;

#define B_   2
#define T_   4096
#define CH_  384
#define HID_ 96
#define DI_  192
#define DS_  16
#define DR_  6
#define EPS_ 1e-5f
#define KC1  (3*CH_)      // conv1 im2col K = 1152
#define LDBC 48           // padded DBC leading dim

__device__ __forceinline__ float sigmoidf_(float x){ return 1.f/(1.f+__expf(-x)); }
__device__ __forceinline__ float siluf_(float x){ return x*sigmoidf_(x); }
__device__ __forceinline__ float softplusf_(float x){ return (x>20.f)?x:log1pf(__expf(x)); }
__device__ __forceinline__ float geluf_(float x){ return 0.5f*x*(1.f+erff(x*0.7071067811865475f)); }

// ---------------------------------------------------------------------------
// Generic NT GEMM via V_WMMA_F32_16X16X4_F32, guard-free hot loop.
// Requirements (enforced by callers via padding): M%16==0, N%(16*NTW)==0,
// K%4==0, all pointers/leading dims 8-byte aligned (even float counts).
//   Out[m,n] = sum_k A[m,k]*Bw[n,k]  (+bias[n], optional softplus)
// One wave computes a 16 x (16*NTW) tile: A fragment loaded once per K-step,
// NTW back-to-back WMMAs. Optional time-reversal of A rows for the backward
// mamba. Tile id is wave-uniform -> EXEC all-ones at WMMA.
// ---------------------------------------------------------------------------
template<int NTW>
__global__ void gemm_nt_wmma(const float* __restrict__ A, int lda, int rev,
                             const float* __restrict__ Bw,
                             const float* __restrict__ bias, int act,
                             float* __restrict__ Out, int ldo,
                             int M, int N, int K)
{
    const int lane = threadIdx.x & 31;
    const int wv   = threadIdx.x >> 5;
    const int groupsN = N / (16*NTW);
    long g = (long)blockIdx.x * (blockDim.x >> 5) + wv;
    if (g >= (long)(M/16) * groupsN) return;            // wave-uniform
    const int tm = (int)(g / groupsN);
    const int tg = (int)(g % groupsN);
    const int half = lane >> 4, lr = lane & 15;

    const int m = tm*16 + lr;
    long mA = m;
    if (rev) { int b = m / T_; int t = m - b*T_; mA = (long)b*T_ + (T_-1-t); }
    const float* Arow = A + mA*lda;

    const float* Brow[NTW];
#pragma unroll
    for (int j = 0; j < NTW; ++j)
        Brow[j] = Bw + (long)((tg*NTW + j)*16 + lr) * K;

    v8f acc[NTW];
#pragma unroll
    for (int j = 0; j < NTW; ++j) acc[j] = (v8f){0.f,0.f,0.f,0.f,0.f,0.f,0.f,0.f};

    for (int k0 = 0; k0 < K; k0 += 4) {
        const int ka = k0 + half*2;
        v2f av = *(const v2f*)(Arow + ka);
#pragma unroll
        for (int j = 0; j < NTW; ++j) {
            v2f bv = *(const v2f*)(Brow[j] + ka);
            acc[j] = __builtin_amdgcn_wmma_f32_16x16x4_f32(false, av, false, bv,
                                                           (short)0, acc[j], false, false);
        }
    }
#pragma unroll
    for (int j = 0; j < NTW; ++j) {
        const int nc = (tg*NTW + j)*16 + lr;
        float bz = bias ? bias[nc] : 0.f;
#pragma unroll
        for (int r = 0; r < 8; ++r) {
            int mr = tm*16 + half*8 + r;                // D layout: VGPR r -> row r / r+8
            float v = acc[j][r] + bz;
            if (act == 1) v = softplusf_(v);
            Out[(long)mr*ldo + nc] = v;
        }
    }
}

// ---------------------------------------------------------------------------
// conv1 (dilated, 'same', K=3) as implicit-im2col WMMA GEMM, NTW=2 (16x32
// tile per wave). A read from time-major transposed input XT (B,T,CH) with
// clamp+mask boundary handling (v_cndmask, no EXEC divergence). B read from
// kt-major packed weights. Also accumulates per-batch GN1 sum / sumsq.
// ---------------------------------------------------------------------------
__global__ void conv1_gemm_stats(const float* __restrict__ XT,   // (B,T,CH)
                                 const float* __restrict__ Wp,   // (HID, 3*CH) kt-major
                                 const float* __restrict__ bias, // (HID)
                                 float* __restrict__ HT,         // (B,T,HID)
                                 float* __restrict__ stats, int dil)
{
    const int lane = threadIdx.x & 31;
    const int wv   = threadIdx.x >> 5;
    const int groupsN = HID_/32;                        // 3
    long g = (long)blockIdx.x*(blockDim.x>>5) + wv;
    if (g >= (long)(B_*T_/16)*groupsN) return;          // wave-uniform
    const int tm = (int)(g / groupsN);
    const int tg = (int)(g % groupsN);
    const int half = lane >> 4, lr = lane & 15;

    const int m = tm*16 + lr;
    const int b = m / T_, t = m - b*T_;
    const int n0 = tg*32 + lr;
    const int n1 = n0 + 16;

    v8f acc0 = (v8f){0.f,0.f,0.f,0.f,0.f,0.f,0.f,0.f};
    v8f acc1 = acc0;
    for (int kt = 0; kt < 3; ++kt) {
        int tt  = t + (kt-1)*dil;
        int ttc = min(max(tt, 0), T_-1);
        float msk = (tt == ttc) ? 1.f : 0.f;            // in-range mask
        const float* Arow = XT + ((long)(b*T_ + ttc))*CH_;
        const float* B0 = Wp + (long)n0*KC1 + kt*CH_;
        const float* B1 = Wp + (long)n1*KC1 + kt*CH_;
        for (int i = 0; i < CH_; i += 4) {
            const int ia = i + half*2;
            v2f av = *(const v2f*)(Arow + ia);
            av.x *= msk; av.y *= msk;
            v2f b0 = *(const v2f*)(B0 + ia);
            v2f b1 = *(const v2f*)(B1 + ia);
            acc0 = __builtin_amdgcn_wmma_f32_16x16x4_f32(false, av, false, b0,
                                                         (short)0, acc0, false, false);
            acc1 = __builtin_amdgcn_wmma_f32_16x16x4_f32(false, av, false, b1,
                                                         (short)0, acc1, false, false);
        }
    }
    float ls = 0.f, ls2 = 0.f;
    const float bz0 = bias[n0], bz1 = bias[n1];
#pragma unroll
    for (int r = 0; r < 8; ++r) {
        int mr = tm*16 + half*8 + r;
        float v0 = acc0[r] + bz0;
        float v1 = acc1[r] + bz1;
        HT[(long)mr*HID_ + n0] = v0;
        HT[(long)mr*HID_ + n1] = v1;
        ls += v0 + v1; ls2 += v0*v0 + v1*v1;
    }
    // all 32 lanes of this wave belong to the same batch b
    for (int off = 16; off >= 1; off >>= 1) {
        ls  += __shfl_xor(ls,  off, 32);
        ls2 += __shfl_xor(ls2, off, 32);
    }
    if (lane == 0) {
        atomicAdd(&stats[b*2  ], ls);
        atomicAdd(&stats[b*2+1], ls2);
    }
}

// --------------------------- small prep kernels ----------------------------
__global__ void zero_stats(float* __restrict__ s)
{
    if (threadIdx.x < 8) s[threadIdx.x] = 0.f;
}

__global__ void transpose_bct_btc(const float* __restrict__ Xin, float* __restrict__ XT)
{
    long idx = (long)blockIdx.x*blockDim.x + threadIdx.x;
    if (idx >= (long)B_*T_*CH_) return;
    int c  = (int)(idx % CH_);
    long m = idx / CH_;
    int b = (int)(m / T_), t = (int)(m % T_);
    XT[idx] = Xin[((long)(b*CH_+c))*T_ + t];
}

__global__ void pack_c1w(const float* __restrict__ W, float* __restrict__ Wp)
{   // Wp[n*1152 + kt*384 + i] = W[n*1152 + i*3 + kt]
    int idx = blockIdx.x*blockDim.x + threadIdx.x;
    if (idx >= HID_*KC1) return;
    int n = idx / KC1, kk = idx % KC1;
    int kt = kk / CH_, i = kk % CH_;
    Wp[idx] = W[n*KC1 + i*3 + kt];
}

__global__ void pack_xproj(const float* __restrict__ W, float* __restrict__ Wp)
{   // (38,192) -> (48,192) zero-padded rows
    int idx = blockIdx.x*blockDim.x + threadIdx.x;
    if (idx >= 48*DI_) return;
    int n = idx / DI_, k = idx % DI_;
    Wp[idx] = (n < 38) ? W[n*DI_ + k] : 0.f;
}

__global__ void pack_dtw(const float* __restrict__ W, float* __restrict__ Wp)
{   // (192,6) -> (192,8) zero-padded K
    int idx = blockIdx.x*blockDim.x + threadIdx.x;
    if (idx >= DI_*8) return;
    int n = idx / 8, k = idx % 8;
    Wp[idx] = (k < 6) ? W[n*6 + k] : 0.f;
}

// --------------------------- elementwise kernels ---------------------------
__global__ void gn1_gelu(float* __restrict__ HT, const float* __restrict__ w,
                         const float* __restrict__ gb, const float* __restrict__ stats)
{
    long idx = (long)blockIdx.x*blockDim.x + threadIdx.x;
    if (idx >= (long)B_*T_*HID_) return;
    int o  = (int)(idx % HID_);
    long m = idx / HID_;
    int b  = (int)(m / T_);
    const float Nn = (float)(T_*HID_);
    float mu  = stats[b*2] / Nn;
    float var = stats[b*2+1] / Nn - mu*mu;
    float h = HT[idx];
    h = (h - mu) * rsqrtf(var + EPS_) * w[o] + gb[o];
    HT[idx] = geluf_(h);
}

// depthwise causal conv (k=4, pad (3,0)) over xm = XZ[...,0:DI], + SiLU
__global__ void dwconv_silu(const float* __restrict__ XZ, const float* __restrict__ W4,
                            const float* __restrict__ bias, float* __restrict__ XC)
{
    long idx = (long)blockIdx.x*blockDim.x + threadIdx.x;
    if (idx >= (long)B_*T_*DI_) return;
    int c  = (int)(idx % DI_);
    long m = idx / DI_;
    int b  = (int)(m / T_), t = (int)(m % T_);
    float s = bias[c];
#pragma unroll
    for (int k = 0; k < 4; ++k) {
        int tt = t - 3 + k;
        if (tt >= 0) s += XZ[((long)(b*T_+tt))*(2*DI_) + c] * W4[c*4 + k];
    }
    XC[idx] = siluf_(s);
}

// ---------------------------------------------------------------------------
// Selective scan: one wave handles 2 channels; lane = (channel-half, state s).
// h[s] <- exp(dt*A[c,s])*h[s] + dt*xm*B[t,s];  y = (sum_s h*C[t,s] + xm*D)*silu(z)
// ---------------------------------------------------------------------------
__global__ void mamba_scan(const float* __restrict__ DT,  const float* __restrict__ XC,
                           const float* __restrict__ DBC, const float* __restrict__ XZ,
                           const float* __restrict__ A_log, const float* __restrict__ Dp,
                           float* __restrict__ YD)
{
    const int lane = threadIdx.x & 31;
    const int wv   = threadIdx.x >> 5;
    int wg = blockIdx.x*(blockDim.x>>5) + wv;
    if (wg >= B_*DI_/2) return;
    const int half = lane >> 4, s = lane & 15;
    int pair = wg*2 + half;
    int b = pair / DI_, c = pair % DI_;

    const float Acs = -__expf(A_log[c*DS_ + s]);   // A = -exp(A_log)
    const float Dc  = Dp[c];
    float h = 0.f;
    const long base = (long)b * T_;
    for (int t = 0; t < T_; ++t) {
        long m = base + t;
        float dtv = DT[m*DI_ + c];
        float xmv = XC[m*DI_ + c];
        float Bv  = DBC[m*LDBC + DR_ + s];
        float Cv  = DBC[m*LDBC + DR_ + DS_ + s];
        float dA  = __expf(dtv * Acs);
        h = dA*h + (dtv*xmv)*Bv;
        float p = h * Cv;
        p += __shfl_xor(p, 1, 32);
        p += __shfl_xor(p, 2, 32);
        p += __shfl_xor(p, 4, 32);
        p += __shfl_xor(p, 8, 32);
        if (s == 0) {
            float zv = XZ[m*(2*DI_) + DI_ + c];
            YD[m*DI_ + c] = (p + xmv*Dc) * siluf_(zv);
        }
    }
}

// GLU: v = a*sigmoid(g) in-place over the 'a' half of Y2, + per-batch GN2 stats
__global__ void glu_stats(float* __restrict__ Y2, float* __restrict__ stats)
{
    long idx = (long)blockIdx.x*blockDim.x + threadIdx.x;
    float v = 0.f; int b = 0;
    bool ok = idx < (long)B_*T_*CH_;
    if (ok) {
        int c  = (int)(idx % CH_);
        long m = idx / CH_;
        b = (int)(m / T_);
        float a = Y2[m*768 + c];
        float gg = Y2[m*768 + 384 + c];
        v = a * sigmoidf_(gg);
        Y2[m*768 + c] = v;
    }
    float ls = v, ls2 = v*v;
    for (int off = 16; off >= 1; off >>= 1) {
        ls  += __shfl_xor(ls,  off, 32);
        ls2 += __shfl_xor(ls2, off, 32);
    }
    if ((threadIdx.x & 31) == 0 && ok) {
        atomicAdd(&stats[4 + b*2], ls);
        atomicAdd(&stats[5 + b*2], ls2);
    }
}

__global__ void final_add(const float* __restrict__ Xin, const float* __restrict__ Y2,
                          const float* __restrict__ stats, const float* __restrict__ gw,
                          const float* __restrict__ gb, const float* __restrict__ scale,
                          float* __restrict__ Xout)
{
    long idx = (long)blockIdx.x*blockDim.x + threadIdx.x;
    if (idx >= (long)B_*CH_*T_) return;
    int t  = (int)(idx % T_);
    long r = idx / T_;
    int c  = (int)(r % CH_);
    int b  = (int)(r / CH_);
    const float Nn = (float)(CH_*T_);
    float mu  = stats[4 + b*2] / Nn;
    float var = stats[5 + b*2] / Nn - mu*mu;
    float v = Y2[((long)(b*T_+t))*768 + c];
    v = (v - mu) * rsqrtf(var + EPS_) * gw[c] + gb[c];
    Xout[idx] = Xin[idx] + scale[c]*v;
}

// ---------------------------------------------------------------------------
// Host orchestration. Input flattening (dict insertion order):
//   0: x
//   per layer (27): c1w c1b gn1w gn1b | mf(9) | mb(9) | c2w c2b gn2w gn2b scale
//   mamba(9): in_w conv_w conv_b xproj_w dt_w dt_b A_log D out_w
// ---------------------------------------------------------------------------
extern "C" void kernel_launch(void* const* d_in, const int* in_sizes, int n_in,
                              void* d_out, int out_size, void* d_ws, size_t ws_size,
                              hipStream_t stream)
{
    (void)in_sizes; (void)n_in; (void)out_size; (void)ws_size;
    const float* X0 = (const float*)d_in[0];
    float* ws = (float*)d_ws;

    // workspace layout (floats), all offsets even  (~80.7 MB total)
    float* XBUF = ws;                 // (B,CH,T)      3,145,728
    float* HT   = XBUF + 3145728;     // (B,T,HID)       786,432
    float* XZ   = HT   + 786432;      // (B,T,2*DI)    3,145,728
    float* XC   = XZ   + 3145728;     // (B,T,DI)      1,572,864
    float* DBC  = XC   + 1572864;     // (B,T,48)        393,216
    float* DT   = DBC  + 393216;      // (B,T,DI)      1,572,864
    float* YD   = DT   + 1572864;     // (B,T,DI)      1,572,864
    float* YCAT = YD   + 1572864;     // (B,T,2*HID)   1,572,864
    float* Y2   = YCAT + 1572864;     // (B,T,768)     6,291,456
    float* STATS= Y2   + 6291456;     // 8
    float* C1WP = STATS + 8;          // (HID,1152)      110,592
    float* XPW  = C1WP + 110592;      // (48,192)          9,216
    float* DTW  = XPW  + 9216;        // (192,8)           1,536
    float* XT   = Y2;                 // (B,T,CH) aliases Y2 (conv1 phase only)

    const int M = B_*T_;

    for (int L = 0; L < 2; ++L) {
        void* const* P = d_in + 1 + L*27;
        const float* c1w  = (const float*)P[0];
        const float* c1b  = (const float*)P[1];
        const float* gn1w = (const float*)P[2];
        const float* gn1b = (const float*)P[3];
        const float* c2w  = (const float*)P[22];
        const float* c2b  = (const float*)P[23];
        const float* gn2w = (const float*)P[24];
        const float* gn2b = (const float*)P[25];
        const float* lscale = (const float*)P[26];
        const float* Xin = (L == 0) ? X0 : XBUF;
        float* Xout = (L == 0) ? XBUF : (float*)d_out;
        const int dil = (L == 0) ? 1 : 2;

        zero_stats<<<1, 32, 0, stream>>>(STATS);
        pack_c1w<<<(HID_*KC1 + 255)/256, 256, 0, stream>>>(c1w, C1WP);
        transpose_bct_btc<<<(int)(((long)M*CH_ + 255)/256), 256, 0, stream>>>(Xin, XT);
        {
            int groups = (M/16) * (HID_/32);          // NTW=2
            conv1_gemm_stats<<<(groups + 3)/4, 128, 0, stream>>>(
                XT, C1WP, c1b, HT, STATS, dil);
        }
        gn1_gelu<<<(int)(((long)M*HID_ + 255)/256), 256, 0, stream>>>(HT, gn1w, gn1b, STATS);

        for (int dir = 0; dir < 2; ++dir) {   // 0: forward, 1: time-reversed
            void* const* MP = P + 4 + dir*9;
            const float* in_w    = (const float*)MP[0];
            const float* conv_w  = (const float*)MP[1];
            const float* conv_b  = (const float*)MP[2];
            const float* xproj_w = (const float*)MP[3];
            const float* dt_w    = (const float*)MP[4];
            const float* dt_b    = (const float*)MP[5];
            const float* A_log   = (const float*)MP[6];
            const float* Dp      = (const float*)MP[7];
            const float* out_w   = (const float*)MP[8];

            pack_xproj<<<(48*DI_ + 255)/256, 256, 0, stream>>>(xproj_w, XPW);
            pack_dtw<<<(DI_*8 + 255)/256, 256, 0, stream>>>(dt_w, DTW);

            // xz = ht @ in_w^T   (N=384, NTW=4; reversal folded into A rows)
            gemm_nt_wmma<4><<<((M/16)*(2*DI_/64) + 3)/4, 128, 0, stream>>>(
                HT, HID_, dir, in_w, nullptr, 0, XZ, 2*DI_, M, 2*DI_, HID_);
            dwconv_silu<<<(int)(((long)M*DI_ + 255)/256), 256, 0, stream>>>(
                XZ, conv_w, conv_b, XC);
            // dbc = xm_c @ xproj_w^T   (N=48 padded, NTW=1)
            gemm_nt_wmma<1><<<((M/16)*(48/16) + 3)/4, 128, 0, stream>>>(
                XC, DI_, 0, XPW, nullptr, 0, DBC, LDBC, M, 48, DI_);
            // dt = softplus(dbc[:, :8] @ dtw^T + dt_b)   (K padded to 8, NTW=4)
            gemm_nt_wmma<4><<<((M/16)*(DI_/64) + 3)/4, 128, 0, stream>>>(
                DBC, LDBC, 0, DTW, dt_b, 1, DT, DI_, M, DI_, 8);
            {
                int waves = B_*DI_/2;
                mamba_scan<<<(waves + 3)/4, 128, 0, stream>>>(
                    DT, XC, DBC, XZ, A_log, Dp, YD);
            }
            // out-proj into its half of the concat buffer (N=96, NTW=2)
            gemm_nt_wmma<2><<<((M/16)*(HID_/32) + 3)/4, 128, 0, stream>>>(
                YD, DI_, 0, out_w, nullptr, 0, YCAT + dir*HID_, 2*HID_, M, HID_, DI_);
        }

        // c2 (1x1 conv): (B,T,192) @ (768,192)^T + c2b   (N=768, NTW=4)
        gemm_nt_wmma<4><<<((M/16)*(768/64) + 3)/4, 128, 0, stream>>>(
            YCAT, 2*HID_, 0, c2w, c2b, 0, Y2, 768, M, 768, 2*HID_);
        glu_stats<<<(int)(((long)M*CH_ + 255)/256), 256, 0, stream>>>(Y2, STATS);
        final_add<<<(int)(((long)B_*CH_*T_ + 255)/256), 256, 0, stream>>>(
            Xin, Y2, STATS, gn2w, gn2b, lscale, Xout);
    }
}